// MemristiveLinear_3702261809494
// MI455X (gfx1250) — compile-verified
//
#include <hip/hip_runtime.h>
#include <stdint.h>

// y = x @ w + b  (the memristive conductance mapping cancels exactly:
//  g_pos - g_neg = g_eff, and the K_V * k_cond scaling divides back out)

typedef __attribute__((ext_vector_type(2))) float v2f;
typedef __attribute__((ext_vector_type(8))) float v8f;

#define B_DIM 512   // batch rows (M)
#define K_DIM 512   // inner dim
#define N_DIM 512   // output cols

#define BM 64       // block tile M
#define BN 64       // block tile N
#define BK 32       // K tile
#define LDP 36      // LDS row pitch (floats): 144B rows, 16B-aligned, conflict-free float2 reads

// Generic pointers to LDS carry the LDS byte offset in their low 32 bits
__device__ __forceinline__ uint32_t lds_off(const void* p) {
    return (uint32_t)(uintptr_t)p;
}

// CDNA5 async global->LDS copy, tracked by ASYNCcnt (gfx1250 VGLOBAL encoding).
__device__ __forceinline__ void async_b128(uint32_t dst_lds, const void* src) {
    asm volatile("global_load_async_to_lds_b128 %0, %1, off"
                 :: "v"(dst_lds), "v"((uint64_t)(uintptr_t)src)
                 : "memory");
}

__device__ __forceinline__ void wait_async0() {
    asm volatile("s_wait_asynccnt 0x0" ::: "memory");
}

__global__ __launch_bounds__(128)
void memristive_linear_wmma_f32(const float* __restrict__ x,
                                const float* __restrict__ w,
                                const float* __restrict__ bias,
                                float* __restrict__ y) {
    __shared__ __align__(16) float As [2][BM][LDP];  // [m][k]  (async straight copy of x tile)
    __shared__ __align__(16) float Bsh[2][BN][LDP];  // [n][k]  (sync transposed copy of w tile)

    const int tid  = threadIdx.x;
    const int lane = tid & 31;           // wave32 lane
    const int wv   = tid >> 5;           // wave id 0..3
    const int wm   = wv >> 1;            // 0..1
    const int wn   = wv & 1;             // 0..1

    const int row_base = blockIdx.y * BM;
    const int col_base = blockIdx.x * BN;

    // fp32 WMMA 16x16x4 fragment indexing:
    //   A: lane holds A[m][k0], A[m][k0+1]   m = lane%16, k0 = (lane/16)*2
    //   B: lane holds B[k0][n], B[k0+1][n]   n = lane%16
    //   D: VGPR v, lanes 0-15 -> D[v][n]; lanes 16-31 -> D[v+8][n]
    const int lm = lane & 15;
    const int kp = (lane >> 4) * 2;

    // staging coordinates (128 threads)
    const int ar  = tid >> 3;            // 0..15 (A rows ar, +16, +32, +48; 16B per async op)
    const int ac  = (tid & 7) * 4;       // 0..28
    const int bkr = tid >> 4;            // 0..7  (B k-rows bkr, +8, +16, +24)
    const int bnc = (tid & 15) * 4;      // 0..60

    // A tile via CDNA5 async copies (no VGPR round-trip, ASYNCcnt-tracked)
    auto stage_a = [&](int buf, int kk) {
#pragma unroll
        for (int i = 0; i < 4; ++i) {
            const int r = ar + i * 16;
            async_b128(lds_off(&As[buf][r][ac]),
                       &x[(row_base + r) * K_DIM + kk + ac]);
        }
    };
    // B tile transposed in LDS so fragments are contiguous float2 (no packing movs)
    auto stage_b = [&](int buf, int kk) {
#pragma unroll
        for (int i = 0; i < 4; ++i) {
            const int kr = bkr + i * 8;
            const float4 vb = *(const float4*)&w[(kk + kr) * N_DIM + col_base + bnc];
            Bsh[buf][bnc + 0][kr] = vb.x;
            Bsh[buf][bnc + 1][kr] = vb.y;
            Bsh[buf][bnc + 2][kr] = vb.z;
            Bsh[buf][bnc + 3][kr] = vb.w;
        }
    };

    // 2x2 register blocking: each wave owns a 32x32 output patch
    v8f acc00 = {}, acc01 = {}, acc10 = {}, acc11 = {};

    // ---- prologue: land tile 0 ----
    stage_a(0, 0);
    stage_b(0, 0);
    wait_async0();
    __syncthreads();

    int buf = 0;
    for (int kk = 0; kk < K_DIM; kk += BK) {
        // stage the next K tile into the other buffer while computing this one
        if (kk + BK < K_DIM) {
            stage_a(buf ^ 1, kk + BK);
            stage_b(buf ^ 1, kk + BK);
        }

        // ---- 32 x v_wmma_f32_16x16x4_f32 from the current buffer ----
        const float* ap0 = &As [buf][wm * 32 + lm][kp];
        const float* ap1 = ap0 + 16 * LDP;
        const float* bp0 = &Bsh[buf][wn * 32 + lm][kp];
        const float* bp1 = bp0 + 16 * LDP;
#pragma unroll
        for (int ki = 0; ki < BK / 4; ++ki) {
            const v2f a0 = *(const v2f*)(ap0 + ki * 4);
            const v2f a1 = *(const v2f*)(ap1 + ki * 4);
            const v2f b0 = *(const v2f*)(bp0 + ki * 4);
            const v2f b1 = *(const v2f*)(bp1 + ki * 4);
            acc00 = __builtin_amdgcn_wmma_f32_16x16x4_f32(false, a0, false, b0, (short)0, acc00, false, false);
            acc01 = __builtin_amdgcn_wmma_f32_16x16x4_f32(false, a0, false, b1, (short)0, acc01, false, false);
            acc10 = __builtin_amdgcn_wmma_f32_16x16x4_f32(false, a1, false, b0, (short)0, acc10, false, false);
            acc11 = __builtin_amdgcn_wmma_f32_16x16x4_f32(false, a1, false, b1, (short)0, acc11, false, false);
        }

        wait_async0();      // my async A copies for buf^1 have landed in LDS
        __syncthreads();    // + compiler-inserted dscnt wait covers the sync B stores
        buf ^= 1;
    }

    // ---- epilogue: add bias, store per the C/D VGPR layout ----
    const int n0 = col_base + wn * 32 + lm;
    const int n1 = n0 + 16;
    const float bv0 = bias[n0];
    const float bv1 = bias[n1];
    const int m0 = row_base + wm * 32 + (lane >> 4) * 8;
#pragma unroll
    for (int v = 0; v < 8; ++v) {
        y[(m0 + v) * N_DIM + n0]      = acc00[v] + bv0;
        y[(m0 + v) * N_DIM + n1]      = acc01[v] + bv1;
        y[(m0 + 16 + v) * N_DIM + n0] = acc10[v] + bv0;
        y[(m0 + 16 + v) * N_DIM + n1] = acc11[v] + bv1;
    }
}

extern "C" void kernel_launch(void* const* d_in, const int* in_sizes, int n_in,
                              void* d_out, int out_size, void* d_ws, size_t ws_size,
                              hipStream_t stream) {
    (void)in_sizes; (void)n_in; (void)out_size; (void)d_ws; (void)ws_size;
    const float* x    = (const float*)d_in[0];
    const float* w    = (const float*)d_in[1];
    const float* bias = (const float*)d_in[2];
    float* y = (float*)d_out;

    dim3 grid(N_DIM / BN, B_DIM / BM);   // (8, 8)
    dim3 block(128);                     // 4 wave32s
    memristive_linear_wmma_f32<<<grid, block, 0, stream>>>(x, w, bias, y);
}